// TokenProcessor_50354196579047
// MI455X (gfx1250) — compile-verified
//
#include <hip/hip_runtime.h>
#include <cfloat>

typedef __attribute__((ext_vector_type(2))) float v2f;
typedef __attribute__((ext_vector_type(8))) float v8f;

#define ROWS    16
#define KTOK    2048
#define THREADS 512
#define WAVES   16
#define TILES_PER_WAVE (KTOK / 16 / WAVES)   // 8

__launch_bounds__(THREADS, 1)
__global__ void token_sample_kernel(const float* __restrict__ traj_pos,   // (N,3,2)
                                    const float* __restrict__ traj_theta, // (N)
                                    const float* __restrict__ bmat,       // (K,3,2) == (K,6)
                                    const float* __restrict__ src,        // (K,22)
                                    const float* __restrict__ gum,        // (N,32)
                                    float* __restrict__ out,              // (N,22)
                                    int N, int K)
{
    __shared__ float sA[ROWS * 8];       // rotated A rows, padded K-dim 6 -> 8 (pads = 0)
    __shared__ float sAn[ROWS];          // ||a||^2
    __shared__ float sBn[KTOK];          // ||b||^2
    __shared__ float sDist[ROWS * KTOK]; // 128 KB distance panel

    const int tid  = threadIdx.x;
    const int lane = tid & 31;
    const int wave = tid >> 5;
    const int rowBase = blockIdx.x * ROWS;

    // ---- Phase 0a: build rotated A rows + anorm (threads 0..15, one row each)
    if (tid < ROWS) {
        int n = rowBase + tid;
        if (n < N) {
            float th = traj_theta[n];
            float c = cosf(th), s = sinf(th);
            float px0 = traj_pos[n * 6 + 0];
            float py0 = traj_pos[n * 6 + 1];
            float norm = 0.0f;
            #pragma unroll
            for (int t = 0; t < 3; ++t) {
                float dx = traj_pos[n * 6 + 2 * t + 0] - px0;
                float dy = traj_pos[n * 6 + 2 * t + 1] - py0;
                float lx =  c * dx + s * dy;
                float ly = -s * dx + c * dy;
                sA[tid * 8 + 2 * t + 0] = lx;
                sA[tid * 8 + 2 * t + 1] = ly;
                norm += lx * lx + ly * ly;
            }
            sA[tid * 8 + 6] = 0.0f;
            sA[tid * 8 + 7] = 0.0f;
            sAn[tid] = norm;
        } else {
            #pragma unroll
            for (int j = 0; j < 8; ++j) sA[tid * 8 + j] = 0.0f;
            sAn[tid] = 0.0f;
        }
    }
    // ---- Phase 0b: token norms (all threads cooperate)
    for (int k = tid; k < K && k < KTOK; k += THREADS) {
        float nb = 0.0f;
        #pragma unroll
        for (int j = 0; j < 6; ++j) { float v = bmat[k * 6 + j]; nb += v * v; }
        sBn[k] = nb;
    }
    __syncthreads();

    // ---- Phase 1: distance panel via WMMA f32 16x16x4 (two k-steps, pad 6->8)
    // Constant trip count + unconditional loads keep EXEC all-1s around WMMA.
    {
        const int khalf = lane >> 4;   // K-half selector per ISA 32-bit A/B layout
        const int m0    = lane & 15;

        // A fragment (16x4 f32): lanes 0-15 -> K0/K1, lanes 16-31 -> K2/K3
        v2f a0, a1;
        a0.x = sA[m0 * 8 + (khalf ? 2 : 0)];
        a0.y = sA[m0 * 8 + (khalf ? 3 : 1)];
        a1.x = sA[m0 * 8 + (khalf ? 6 : 4)];   // pads at 6,7 are zero
        a1.y = sA[m0 * 8 + (khalf ? 7 : 5)];

        #pragma unroll
        for (int i = 0; i < TILES_PER_WAVE; ++i) {
            int tc  = wave + i * WAVES;
            int col = tc * 16 + m0;
            // B fragment (4x16). For the second k-step, khalf=1 lanes cover the
            // K=6,7 zero pads: A is zero there, so any finite in-bounds B value
            // works -> unconditional loads, no exec-mask branches.
            v2f b0, b1;
            b0.x = bmat[col * 6 + (khalf ? 2 : 0)];
            b0.y = bmat[col * 6 + (khalf ? 3 : 1)];
            b1.x = bmat[col * 6 + (khalf ? 2 : 4)];   // khalf=1: don't-care (A pad = 0)
            b1.y = bmat[col * 6 + (khalf ? 3 : 5)];   // khalf=1: don't-care (A pad = 0)

            v8f acc = {};
            acc = __builtin_amdgcn_wmma_f32_16x16x4_f32(false, a0, false, b0,
                                                        (short)0, acc, false, false);
            acc = __builtin_amdgcn_wmma_f32_16x16x4_f32(false, a1, false, b1,
                                                        (short)0, acc, false, false);

            float bn = sBn[col];
            // D layout: VGPR v -> row M = v + 8*khalf, col = lane&15 within tile
            #pragma unroll
            for (int v = 0; v < 8; ++v) {
                int row = v + 8 * khalf;
                sDist[row * KTOK + col] = sAn[row] + bn - 2.0f * acc[v];
            }
        }
    }
    __syncthreads();

    // ---- Phase 2: per-row top-32 extraction fused with Gumbel argmax + gather
    {
        const int r = wave;              // one wave per row
        const int n = rowBase + r;
        if (n < N) {
            const int per = KTOK / 32;   // 64 strided entries per lane

            // local min over this lane's strided slice
            float lmin = FLT_MAX; int lidx = lane;
            for (int t = 0; t < per; ++t) {
                int j = lane + 32 * t;
                float v = sDist[r * KTOK + j];
                if (v < lmin) { lmin = v; lidx = j; }
            }

            float best = -FLT_MAX;
            int bestTok = 0;
            for (int it = 0; it < 32; ++it) {
                // wave-wide (val,idx) min via butterfly; ties -> lower index
                float gvv = lmin; int gi = lidx;
                #pragma unroll
                for (int off = 16; off > 0; off >>= 1) {
                    float ov = __shfl_xor(gvv, off, 32);
                    int   oi = __shfl_xor(gi,  off, 32);
                    if (ov < gvv || (ov == gvv && oi < gi)) { gvv = ov; gi = oi; }
                }
                // owner lane invalidates the extracted entry and rescans its slice
                if (lidx == gi && lmin == gvv) {
                    sDist[r * KTOK + gi] = FLT_MAX;
                    lmin = FLT_MAX; lidx = lane;
                    for (int t = 0; t < per; ++t) {
                        int j = lane + 32 * t;
                        float v = sDist[r * KTOK + j];
                        if (v < lmin) { lmin = v; lidx = j; }
                    }
                }
                // Gumbel-perturbed score for rank slot `it` (TEMP = 1)
                float u = gum[n * 32 + it];
                float g = -logf(-logf(u * (1.0f - 2.0f * 1e-7f) + 1e-7f));
                float score = (-1e-6f - gvv) + g;
                if (score > best) { best = score; bestTok = gi; }
            }

            // gather 22-float token row (all lanes agree on bestTok)
            if (lane < 22) out[n * 22 + lane] = src[bestTok * 22 + lane];
        }
    }
}

extern "C" void kernel_launch(void* const* d_in, const int* in_sizes, int n_in,
                              void* d_out, int out_size, void* d_ws, size_t ws_size,
                              hipStream_t stream) {
    (void)n_in; (void)d_ws; (void)ws_size; (void)out_size;
    const float* traj_pos   = (const float*)d_in[0];
    const float* traj_theta = (const float*)d_in[1];
    const float* bmat       = (const float*)d_in[2];
    const float* src        = (const float*)d_in[3];
    const float* gum        = (const float*)d_in[4];
    float* out = (float*)d_out;

    int N = in_sizes[1];          // traj_theta is (N,)
    int K = in_sizes[2] / 6;      // map_token_sample_pt is (K,3,2)

    int blocks = (N + ROWS - 1) / ROWS;
    token_sample_kernel<<<dim3(blocks), dim3(THREADS), 0, stream>>>(
        traj_pos, traj_theta, bmat, src, gum, out, N, K);
}